// RPNProposalSelectorLayer_59622736003500
// MI455X (gfx1250) — compile-verified
//
#include <hip/hip_runtime.h>
#include <cstdint>
#include <cstddef>

#define CLS_THR  0.7f
#define NMS_THR  0.7f
#define MAX_OUT  100
#define MIN_SIZE 3.0f

#define NBLK 32
#define NTHR 256

// ---------------------------------------------------------------------------
// Grid-wide barrier: monotonic-flag, count resets to 0 each episode, so it is
// replay-safe (graph replays reuse d_ws without re-initializing the flag).
// ---------------------------------------------------------------------------
__device__ __forceinline__ void grid_barrier(unsigned int* cnt, unsigned int* flag) {
    __syncthreads();
    if (threadIdx.x == 0) {
        __threadfence();
        unsigned int old = __hip_atomic_load(flag, __ATOMIC_RELAXED, __HIP_MEMORY_SCOPE_AGENT);
        unsigned int tk  = __hip_atomic_fetch_add(cnt, 1u, __ATOMIC_ACQ_REL, __HIP_MEMORY_SCOPE_AGENT);
        if (tk == (unsigned int)(gridDim.x - 1)) {
            __hip_atomic_store(cnt, 0u, __ATOMIC_RELAXED, __HIP_MEMORY_SCOPE_AGENT);
            __hip_atomic_fetch_add(flag, 1u, __ATOMIC_ACQ_REL, __HIP_MEMORY_SCOPE_AGENT);
        } else {
            while (__hip_atomic_load(flag, __ATOMIC_ACQUIRE, __HIP_MEMORY_SCOPE_AGENT) == old) {
                __builtin_amdgcn_s_sleep(2);
            }
        }
        __threadfence();
    }
    __syncthreads();
}

// Per-call workspace init (harness poisons d_ws to 0xAA before timing):
// zero the winner slots and the barrier count. The barrier flag may hold any
// value (it is compared only by inequality).
__global__ void init_ws(unsigned long long* slots, unsigned int* ctrl) {
    int t = threadIdx.x;
    if (t < 128) slots[t] = 0ull;
    if (t == 128) ctrl[0] = 0u;   // barrier count
}

// ---------------------------------------------------------------------------
// Persistent cooperative kernel: threshold + 100-step NMS over the full
// anchor set. active[] is owner-read/owner-write (thread g owns j==g mod G),
// boxes are read straight from the (read-only, L2-resident) proposals input,
// so tie-breaking matches jnp.argmax (lowest original index) bit-exactly.
// ---------------------------------------------------------------------------
__global__ __launch_bounds__(NTHR, 1)
void rpn_nms(const float4* __restrict__ prop,       // [N] yxyx boxes (input)
             const float2* __restrict__ cls,        // [N] (logit0, logit1)
             const int* __restrict__ img_h,
             const int* __restrict__ img_w,
             int N,
             unsigned long long* __restrict__ slots,
             unsigned int* __restrict__ ctrl,
             float* __restrict__ active,            // [N] score or -1
             float* __restrict__ out_boxes,         // [MAX_OUT*4]
             float* __restrict__ out_cls)           // [MAX_OUT]
{
    const int gtid = blockIdx.x * blockDim.x + threadIdx.x;
    const int G    = gridDim.x * blockDim.x;

    // ---- Phase 1: objectness = softmax class-0 prob = sigmoid(l0-l1)
    for (int i = gtid; i < N; i += G) {
        float2 l = cls[i];
        float sc = 1.0f / (1.0f + expf(l.y - l.x));
        active[i] = (sc > CLS_THR) ? sc : -1.0f;
    }

    grid_barrier(&ctrl[0], &ctrl[1]);

    __shared__ unsigned long long s_red[NTHR / 32];
    __shared__ float4 s_wbox;

    bool   wvalid = false;
    float4 wb = make_float4(0.f, 0.f, 0.f, 0.f);
    float  wa = 0.0f;

    const float inv_h = 1.0f / (float)img_h[0];
    const float inv_w = 1.0f / (float)img_w[0];

    // ---- Phase 2: 100 NMS steps. Suppression by winner s-1 is fused into
    // the argmax scan for winner s; the winner removes itself via self-IoU=1.
    for (int s = 0; s < MAX_OUT; ++s) {
        unsigned long long lmax = 0ull;
        for (int j = gtid; j < N; j += G) {
            __builtin_prefetch(&active[j + G], 0, 0);  // global_prefetch_b8
            float a = active[j];
            if (a > 0.0f) {
                if (wvalid) {
                    __builtin_prefetch(&prop[j + G], 0, 0);
                    float4 b = prop[j];
                    float yy1 = fmaxf(wb.x, b.x);
                    float xx1 = fmaxf(wb.y, b.y);
                    float yy2 = fminf(wb.z, b.z);
                    float xx2 = fminf(wb.w, b.w);
                    float inter = fmaxf(yy2 - yy1, 0.0f) * fmaxf(xx2 - xx1, 0.0f);
                    float aj  = (b.z - b.x) * (b.w - b.y);
                    float iou = inter / (aj + wa - inter + 1e-9f);
                    if (iou > NMS_THR) { active[j] = -1.0f; a = -1.0f; }
                }
                if (a > 0.0f) {
                    unsigned long long key =
                        ((unsigned long long)__float_as_uint(a) << 32) |
                        (unsigned long long)(~(unsigned)j);   // lowest index wins ties
                    if (key > lmax) lmax = key;
                }
            }
        }

        // wave32 max-reduce, then per-block fold, then one atomicMax
        for (int off = 16; off > 0; off >>= 1) {
            unsigned long long o = __shfl_xor(lmax, off);
            if (o > lmax) lmax = o;
        }
        if ((threadIdx.x & 31) == 0) s_red[threadIdx.x >> 5] = lmax;
        __syncthreads();
        if (threadIdx.x == 0) {
            unsigned long long bm = s_red[0];
            #pragma unroll
            for (int i = 1; i < NTHR / 32; ++i) if (s_red[i] > bm) bm = s_red[i];
            if (bm) atomicMax(&slots[s], bm);
        }

        grid_barrier(&ctrl[0], &ctrl[1]);

        unsigned long long w =
            __hip_atomic_load(&slots[s], __ATOMIC_RELAXED, __HIP_MEMORY_SCOPE_AGENT);
        unsigned int sb = (unsigned int)(w >> 32);
        wvalid = (sb != 0u);
        float wscore = __uint_as_float(sb);
        unsigned int widx = ~(unsigned int)(w & 0xffffffffull);

        // CDNA5 async DMA: pull the winner box straight into LDS and
        // broadcast it to the whole block (ASYNCcnt path).
        if (threadIdx.x == 0 && wvalid) {
            unsigned lds_box = (unsigned)(uintptr_t)&s_wbox;
            unsigned long long g_box = (unsigned long long)(uintptr_t)&prop[widx];
            asm volatile("global_load_async_to_lds_b128 %0, %1, off"
                         :: "v"(lds_box), "v"(g_box) : "memory");
#if __has_builtin(__builtin_amdgcn_s_wait_asynccnt)
            __builtin_amdgcn_s_wait_asynccnt(0);
#else
            asm volatile("s_wait_asynccnt 0" ::: "memory");
#endif
        }
        __syncthreads();
        if (wvalid) {
            wb = s_wbox;
            wa = (wb.z - wb.x) * (wb.w - wb.y);
        }

        // ---- output record for step s (min-size filter + percentage scale)
        if (gtid == 0) {
            bool keep = wvalid && ((wb.z - wb.x) > MIN_SIZE) && ((wb.w - wb.y) > MIN_SIZE);
            out_boxes[4 * s + 0] = keep ? wb.x * inv_h : 0.0f;
            out_boxes[4 * s + 1] = keep ? wb.y * inv_w : 0.0f;
            out_boxes[4 * s + 2] = keep ? wb.z * inv_h : 0.0f;
            out_boxes[4 * s + 3] = keep ? wb.w * inv_w : 0.0f;
            out_cls[s]           = keep ? wscore : 0.0f;
        }
    }
}

// ---------------------------------------------------------------------------
extern "C" void kernel_launch(void* const* d_in, const int* in_sizes, int n_in,
                              void* d_out, int out_size, void* d_ws, size_t ws_size,
                              hipStream_t stream) {
    const float4* prop = (const float4*)d_in[0];
    const float2* cls  = (const float2*)d_in[1];
    const int* ih      = (const int*)d_in[2];
    const int* iw      = (const int*)d_in[3];
    const int N        = in_sizes[0] / 4;

    // workspace layout: [slots 1KB][ctrl 1KB][active 4N]
    char* ws = (char*)d_ws;
    unsigned long long* slots = (unsigned long long*)ws;
    unsigned int* ctrl        = (unsigned int*)(ws + 1024);
    float* active             = (float*)(ws + 2048);

    float* out_boxes = (float*)d_out;            // [1,100,4] flat
    float* out_cls   = out_boxes + 4 * MAX_OUT;  // [1,100]

    init_ws<<<1, 256, 0, stream>>>(slots, ctrl);
    rpn_nms<<<NBLK, NTHR, 0, stream>>>(prop, cls, ih, iw, N,
                                       slots, ctrl, active,
                                       out_boxes, out_cls);
}